// NeuralAGC_12429635354987
// MI455X (gfx1250) — compile-verified
//
#include <hip/hip_runtime.h>
#include <math.h>

#define T_LEN 131072
#define NB 32
#define NW 64
#define MAX_GAIN_F 1000.0f   // 10^(60/20)

typedef __attribute__((ext_vector_type(2))) float v2f;
typedef __attribute__((ext_vector_type(8))) float v8f;

// ============================================================================
// Kernel 1: power path.  x^2 -> depthwise 64-tap causal conv -> combine -> softplus
// Banded-Toeplitz GEMM on V_WMMA_F32_16X16X4_F32.
//   D[m=time, n=batch] += A_c[m,r4] * B_c[r4,n]
//   A_c[m,r4] = w'[4c + r4 - m]           (w' = conv_w * combine_w, band, 0 outside)
//   B_c[r4,n] = x^2[batch_n][t0 + 4c + r4 - 63]
// grid (T/128, 2): blockIdx.y selects batch half; 8 waves/block, 16 times/wave.
// ============================================================================
#define XS_STRIDE 193   // 193 mod 64 == 1 -> conflict-free column reads

__global__ void k1_power(const float* __restrict__ x,
                         const float* __restrict__ conv_w,
                         const float* __restrict__ combine_w,
                         const float* __restrict__ combine_b,
                         float* __restrict__ p)
{
  __shared__ float wS[2 * 96];            // wS[ch*96 + 16 + k], k in [0,64); zero-padded band
  __shared__ float xs[32 * XS_STRIDE];    // rows: ch*16 + nb, 192-sample time window

  const int tid = threadIdx.x;
  const int bh  = blockIdx.y;             // batch half: batches [16*bh, 16*bh+16)
  const int t0  = blockIdx.x * 128;

  // stage combine-folded weights (padded with zeros on both sides of the band)
  if (tid < 192) {
    int ch = tid / 96, q = tid - ch * 96;
    int k  = q - 16;
    float wv = 0.f;
    if (k >= 0 && k < NW) wv = conv_w[ch * NW + k] * combine_w[ch];
    wS[tid] = wv;
  }
  // stage x^2 window [t0-63, t0+128] (causal zero-pad at t<0)
  for (int i = tid; i < 32 * 192; i += 256) {
    int r = i / 192, j = i - r * 192;
    int ch = r >> 4, nb = r & 15;
    int t  = t0 - 63 + j;
    float v = 0.f;
    if (t >= 0 && t < T_LEN) {
      int batch = bh * 16 + nb;
      v = x[(size_t)(batch * 2 + ch) * T_LEN + t];
    }
    xs[r * XS_STRIDE + j] = v * v;
  }
  __syncthreads();

  const int lane = tid & 31;
  const int wv   = tid >> 5;          // 8 waves -> 8 consecutive 16-time tiles
  const int m    = lane & 15;         // A: row M ; B: column N (batch)
  const int kb   = (lane >> 4) << 1;  // K pair base: 0 (lanes 0-15) or 2 (lanes 16-31)

  v8f acc = {};
  for (int ch = 0; ch < 2; ++ch) {
    const float* wrow = &wS[ch * 96 + 16];
    const float* xrow = &xs[(ch * 16 + m) * XS_STRIDE + 16 * wv];
    #pragma unroll
    for (int c = 0; c < 20; ++c) {
      int k0 = 4 * c + kb - m;        // in [-15, 79); padded array makes OOB reads 0
      v2f a, b;
      a.x = wrow[k0];
      a.y = wrow[k0 + 1];
      int j = 4 * c + kb;             // in [0, 80)
      b.x = xrow[j];
      b.y = xrow[j + 1];
      acc = __builtin_amdgcn_wmma_f32_16x16x4_f32(
          false, a, false, b, (short)0, acc, false, false);
    }
  }

  // D layout: VGPR j -> M = j + 8*(lane>=16), N = lane&15
  const float cb   = combine_b[0];
  const int   n    = lane & 15;
  const int   mhi  = (lane >> 4) * 8;
  float* prow = &p[(size_t)(bh * 16 + n) * T_LEN];
  #pragma unroll
  for (int j2 = 0; j2 < 8; ++j2) {
    int   tt  = t0 + 16 * wv + mhi + j2;
    float pre = acc[j2] + cb;
    float sp  = (pre > 20.f) ? pre : log1pf(__expf(pre));
    prow[tt]  = sp + 1e-8f;
  }
}

// ============================================================================
// Kernel 2: fused double EMA (attack -> gain law -> release) via affine scans.
// One block per batch row; 1024 threads (32 waves), 4 elems/thread, 32 tiles.
// Recurrence y = a*x + d*y  ==  s -> A*s + B with A = d^len.
// ============================================================================
__device__ __forceinline__ void wave_scan(float& A, float& B, int lane)
{
  #pragma unroll
  for (int off = 1; off < 32; off <<= 1) {
    float Ao = __shfl_up(A, off, 32);
    float Bo = __shfl_up(B, off, 32);
    if (lane >= off) { B = fmaf(A, Bo, B); A = A * Ao; }
  }
}

__device__ __forceinline__ void scan_pass(const float x[4], float a, float d, float d4,
                                          float* sh_carry, float* aggA, float* aggB,
                                          float out[4], int lane, int wvi, int tid)
{
  // thread-local affine (length 4, zero-state output)
  float Bl = 0.f;
  #pragma unroll
  for (int i = 0; i < 4; ++i) Bl = fmaf(d, Bl, a * x[i]);
  float Ai = d4, Bi = Bl;
  wave_scan(Ai, Bi, lane);                       // inclusive across the wave
  if (lane == 31) { aggA[wvi] = Ai; aggB[wvi] = Bi; }
  __syncthreads();
  if (wvi == 0) {                                // wave 0 scans the 32 wave aggregates
    float Aw = aggA[lane], Bw = aggB[lane];
    wave_scan(Aw, Bw, lane);
    aggA[lane] = Aw; aggB[lane] = Bw;
  }
  __syncthreads();
  // thread-exclusive prefix within wave
  float Ae = __shfl_up(Ai, 1, 32);
  float Be = __shfl_up(Bi, 1, 32);
  if (lane == 0) { Ae = 1.f; Be = 0.f; }
  float c = *sh_carry;                           // tile-entry state
  float s = (wvi == 0) ? c : fmaf(aggA[wvi - 1], c, aggB[wvi - 1]);
  s = fmaf(Ae, s, Be);
  #pragma unroll
  for (int i = 0; i < 4; ++i) { s = fmaf(d, s, a * x[i]); out[i] = s; }
  __syncthreads();
  if (tid == 0) *sh_carry = fmaf(aggA[31], c, aggB[31]);   // full-tile aggregate
  __syncthreads();
}

__global__ void k2_ema(const float* __restrict__ p,
                       float* __restrict__ gain,
                       const float* __restrict__ la_att,
                       const float* __restrict__ la_rel)
{
  __shared__ float aggA[32], aggB[32];
  __shared__ float sh_c1, sh_c2;
  const int tid  = threadIdx.x;
  const int lane = tid & 31, wvi = tid >> 5;
  const int row  = blockIdx.x;

  const float a1 = 1.f / (1.f + __expf(-la_att[0])), d1 = 1.f - a1;
  const float a2 = 1.f / (1.f + __expf(-la_rel[0])), d2 = 1.f - a2;
  const float d1_4 = d1 * d1 * d1 * d1;
  const float d2_4 = d2 * d2 * d2 * d2;

  const float* pr = p    + (size_t)row * T_LEN;
  float*       gr = gain + (size_t)row * T_LEN;

  if (tid == 0) {                 // scan init: state = first element (so y[0] = x[0])
    float p0 = pr[0];
    sh_c1 = p0;
    sh_c2 = fminf(1.f / (sqrtf(p0) + 1e-8f), MAX_GAIN_F);
  }
  __syncthreads();

  const int TILE = 4096;          // 1024 threads * 4
  for (int tile = 0; tile < T_LEN / TILE; ++tile) {
    int base = tile * TILE + tid * 4;
    float4 x4 = *reinterpret_cast<const float4*>(pr + base);
    float xv[4] = {x4.x, x4.y, x4.z, x4.w};

    float y1[4];
    scan_pass(xv, a1, d1, d1_4, &sh_c1, aggA, aggB, y1, lane, wvi, tid);

    float dg[4];
    #pragma unroll
    for (int i = 0; i < 4; ++i)
      dg[i] = fminf(1.f / (sqrtf(y1[i]) + 1e-8f), MAX_GAIN_F);

    float g[4];
    scan_pass(dg, a2, d2, d2_4, &sh_c2, aggA, aggB, g, lane, wvi, tid);

    *reinterpret_cast<float4*>(gr + base) = make_float4(g[0], g[1], g[2], g[3]);
  }
}

// ============================================================================
// Kernel 3: per-(b,c) sum / sumsq of w = x*gain (enough to derive layernorm stats
// of y = w*dc_w + dc_b analytically). One block per row.
// ============================================================================
__global__ void k3_stats(const float* __restrict__ x, const float* __restrict__ gain,
                         float* __restrict__ sums)
{
  __shared__ float sA[256], sB[256];
  const int row = blockIdx.x;       // b*2 + c
  const int b   = row >> 1;
  const float* xr = x    + (size_t)row * T_LEN;
  const float* gr = gain + (size_t)b   * T_LEN;
  float s = 0.f, s2 = 0.f;
  for (int t = threadIdx.x; t < T_LEN; t += 256) {
    float z = xr[t] * gr[t];
    s += z;
    s2 = fmaf(z, z, s2);
  }
  sA[threadIdx.x] = s; sB[threadIdx.x] = s2;
  __syncthreads();
  for (int off = 128; off > 0; off >>= 1) {
    if (threadIdx.x < off) {
      sA[threadIdx.x] += sA[threadIdx.x + off];
      sB[threadIdx.x] += sB[threadIdx.x + off];
    }
    __syncthreads();
  }
  if (threadIdx.x == 0) { sums[row * 2] = sA[0]; sums[row * 2 + 1] = sB[0]; }
}

// ============================================================================
// Kernel 4: final elementwise  out = LN(x*gain*dc_w + dc_b)*gamma + beta
// ============================================================================
__global__ void k4_out(const float* __restrict__ x, const float* __restrict__ gain,
                       const float* __restrict__ sums,
                       const float* __restrict__ dc_w, const float* __restrict__ dc_b,
                       const float* __restrict__ gma, const float* __restrict__ bta,
                       float* __restrict__ out)
{
  size_t i = (size_t)blockIdx.x * blockDim.x + threadIdx.x;
  int row = (int)(i >> 17);                 // T = 2^17
  int t   = (int)(i & (T_LEN - 1));
  int b   = row >> 1, c = row & 1;

  float w      = x[i] * gain[(size_t)b * T_LEN + t];
  float mean_w = sums[row * 2]     * (1.f / T_LEN);
  float msq_w  = sums[row * 2 + 1] * (1.f / T_LEN);
  float dw = dc_w[c], db = dc_b[c];
  float z   = fmaf(w, dw, db);
  float mu  = fmaf(mean_w, dw, db);
  float var = dw * dw * (msq_w - mean_w * mean_w);
  float r   = rsqrtf(var + 1e-5f);
  out[i] = fmaf((z - mu) * r, gma[c], bta[c]);
}

// ============================================================================
extern "C" void kernel_launch(void* const* d_in, const int* in_sizes, int n_in,
                              void* d_out, int out_size, void* d_ws, size_t ws_size,
                              hipStream_t stream)
{
  const float* x         = (const float*)d_in[0];
  const float* conv_w    = (const float*)d_in[1];
  const float* combine_w = (const float*)d_in[2];
  const float* combine_b = (const float*)d_in[3];
  const float* la_att    = (const float*)d_in[4];
  const float* la_rel    = (const float*)d_in[5];
  const float* dc_w      = (const float*)d_in[6];
  const float* dc_b      = (const float*)d_in[7];
  const float* gma       = (const float*)d_in[8];
  const float* bta       = (const float*)d_in[9];

  float* p    = (float*)d_ws;                    // [32][T]
  float* gain = p    + (size_t)NB * T_LEN;       // [32][T]
  float* sums = gain + (size_t)NB * T_LEN;       // [64][2]
  float* out  = (float*)d_out;

  k1_power<<<dim3(T_LEN / 128, 2), 256, 0, stream>>>(x, conv_w, combine_w, combine_b, p);
  k2_ema  <<<NB, 1024, 0, stream>>>(p, gain, la_att, la_rel);
  k3_stats<<<NB * 2, 256, 0, stream>>>(x, gain, sums);
  k4_out  <<<(NB * 2 * T_LEN) / 256, 256, 0, stream>>>(x, gain, sums, dc_w, dc_b, gma, bta, out);
}